// SphericalHarmonicsEdgeAttrs_2241972929171
// MI455X (gfx1250) — compile-verified
//
#include <hip/hip_runtime.h>
#include <hip/hip_bf16.h>

// ---------------------------------------------------------------------------
// SphericalHarmonics (lmax=3, component norm) over 3.2M edge vectors.
// Memory-bound streaming kernel: async global->LDS staging (CDNA5 TDM-lite
// path, ASYNCcnt) double-buffered against VALU compute, NT b128 stores.
// ---------------------------------------------------------------------------

#define BLOCK 256                     // 8 wave32
#define EPT   4                       // edges per thread
#define TILE  (BLOCK * EPT)           // 1024 edges per tile
#define TILE_F (TILE * 3)             // 3072 floats of input per tile
#define TILE_BYTES (TILE_F * 4)       // 12288 bytes per tile
#define CHUNKS (TILE_BYTES / (BLOCK * 16))  // 3 async b128 issues per thread

typedef float v4f __attribute__((ext_vector_type(4)));

// Issue CHUNKS async b128 copies: global tile -> LDS buffer.
// VDST operand of global_load_async_to_lds_b128 is a VGPR holding the LDS
// byte address; the low 32 bits of a flat pointer into LDS are exactly that
// offset (shared aperture lives in the high bits).
__device__ __forceinline__ void async_tile(const float* __restrict__ in,
                                           float* ldsbuf, long long tile,
                                           int tid) {
  const char* g = (const char*)in + tile * (long long)TILE_BYTES + (long long)tid * 16;
  unsigned l = (unsigned)(unsigned long long)ldsbuf + (unsigned)tid * 16u;
#pragma unroll
  for (int k = 0; k < CHUNKS; ++k) {
    asm volatile("global_load_async_to_lds_b128 %0, %1, off"
                 :
                 : "v"(l + (unsigned)(k * BLOCK * 16)),
                   "v"(g + (long long)(k * BLOCK * 16))
                 : "memory");
  }
}

template <int N>
__device__ __forceinline__ void wait_async() {
  asm volatile("s_wait_asynccnt %0" ::"i"(N) : "memory");
}

// e3nn real SH up to l=3, 'component' normalization, on a (to-normalize) vec.
__device__ __forceinline__ void sh_store(float vx, float vy, float vz,
                                         float* __restrict__ orow) {
  float n2 = vx * vx + vy * vy + vz * vz;
  float inv = 1.0f / sqrtf(fmaxf(n2, 1e-24f));  // == 1/max(norm,1e-12)
  float x = vx * inv, y = vy * inv, z = vz * inv;

  const float s3  = 1.7320508075688772f;   // sqrt(3)
  const float s5  = 2.2360679774997896f;   // sqrt(5)
  const float s15 = 3.8729833462074170f;   // sqrt(15)
  const float c30 = 1.0801234497346435f;   // sqrt(42)/6
  const float c31 = 2.6457513110645907f;   // sqrt(7)
  const float c32 = 1.6201851746019651f;   // sqrt(168)/8
  const float c33 = 1.3228756555322954f;   // sqrt(7)/2

  float y2 = y * y;
  float x2z2 = x * x + z * z;
  float sh20 = s15 * x * z;
  float sh21 = s15 * x * y;
  float sh22 = s5 * (y2 - 0.5f * x2z2);
  float sh23 = s15 * y * z;
  float sh24 = 0.5f * s15 * (z * z - x * x);
  float t4 = 4.0f * y2 - x2z2;

  v4f r0 = {1.0f, s3 * x, s3 * y, s3 * z};
  v4f r1 = {sh20, sh21, sh22, sh23};
  v4f r2 = {sh24,
            c30 * (sh20 * z + sh24 * x),
            c31 * sh20 * y,
            c32 * t4 * x};
  v4f r3 = {c33 * y * (2.0f * y2 - 3.0f * x2z2),
            c32 * z * t4,
            c31 * sh24 * y,
            c30 * (sh24 * z - sh20 * x)};

  v4f* p = (v4f*)orow;
  __builtin_nontemporal_store(r0, p + 0);
  __builtin_nontemporal_store(r1, p + 1);
  __builtin_nontemporal_store(r2, p + 2);
  __builtin_nontemporal_store(r3, p + 3);
}

__global__ void __launch_bounds__(BLOCK)
sh_edge_attrs_kernel(const float* __restrict__ in, float* __restrict__ out,
                     int n) {
  __shared__ __align__(16) float lds[2][TILE_F];
  const int tid = threadIdx.x;
  const int nFull = n / TILE;
  const int G = (int)gridDim.x;

  long long t = blockIdx.x;
  int buf = 0;
  if (t < nFull) async_tile(in, lds[0], t, tid);   // prologue prefetch

  for (; t < nFull; t += G) {
    long long nxt = t + G;
    const bool pre = (nxt < nFull);
    if (pre) async_tile(in, lds[buf ^ 1], nxt, tid);  // overlap next tile
    if (pre) wait_async<CHUNKS>();                    // current tile landed
    else     wait_async<0>();
    __syncthreads();

    const float* lb = lds[buf];
#pragma unroll
    for (int e = 0; e < EPT; ++e) {
      int j = e * BLOCK + tid;                 // stride-3 dword LDS reads:
      float x = lb[3 * j + 0];                 // conflict-free (gcd(3,64)=1)
      float y = lb[3 * j + 1];
      float z = lb[3 * j + 2];
      long long edge = t * (long long)TILE + j;
      sh_store(x, y, z, out + edge * 16);
    }
    __syncthreads();   // all reads of lds[buf] done before it is refilled
    buf ^= 1;
  }

  // Ragged tail (n % TILE edges): direct guarded loads, block 0 only.
  long long tail = (long long)nFull * TILE;
  if (blockIdx.x == 0) {
    for (long long i = tail + tid; i < n; i += BLOCK) {
      float x = in[3 * i + 0];
      float y = in[3 * i + 1];
      float z = in[3 * i + 2];
      sh_store(x, y, z, out + i * 16);
    }
  }
}

extern "C" void kernel_launch(void* const* d_in, const int* in_sizes, int n_in,
                              void* d_out, int out_size, void* d_ws,
                              size_t ws_size, hipStream_t stream) {
  (void)n_in; (void)out_size; (void)d_ws; (void)ws_size;
  const float* in = (const float*)d_in[0];
  float* out = (float*)d_out;
  const int n = in_sizes[0] / 3;   // number of edges

  const int nFull = n / TILE;
  int grid = 1024;
  if (nFull > 0 && grid > nFull) grid = nFull;
  if (nFull == 0) grid = 1;

  sh_edge_attrs_kernel<<<grid, BLOCK, 0, stream>>>(in, out, n);
}